// WDGCNKGEncoder_49340584296884
// MI455X (gfx1250) — compile-verified
//
#include <hip/hip_runtime.h>
#include <hip/hip_bf16.h>

#define HID   64
#define HEADS 4
#define DH    16
#define TOPK  10
#define ALPHA 0.15f
#define SLOPE 0.2f

typedef __attribute__((ext_vector_type(2))) float v2f;
typedef __attribute__((ext_vector_type(8))) float v8f;
typedef __attribute__((ext_vector_type(4))) int   v4i;

#define AS1 __attribute__((address_space(1)))
#define AS3 __attribute__((address_space(3)))

#if __has_builtin(__builtin_amdgcn_global_load_async_to_lds_b128)
#define HAVE_ASYNC_LDS 1
#else
#define HAVE_ASYNC_LDS 0
#endif

#if HAVE_ASYNC_LDS
// 16-byte global -> LDS async copy (GLOBAL_LOAD_ASYNC_TO_LDS_B128, ASYNCcnt)
__device__ __forceinline__ void async_cp16(const void* g, void* l) {
  __builtin_amdgcn_global_load_async_to_lds_b128(
      (AS1 v4i*)g, (AS3 v4i*)l, 0, 0);
}
__device__ __forceinline__ void wait_async0() {
#if __has_builtin(__builtin_amdgcn_s_wait_asynccnt)
  __builtin_amdgcn_s_wait_asynccnt(0);
#else
  asm volatile("s_wait_asynccnt 0x0" ::: "memory");
#endif
}
#endif

// ---------------------------------------------------------------------------
// GEMM: D[M x 64] = act( C0 + A[M x 64] * B[64 x 64] )   (act = ELU optional)
// Block = 256 threads = 8 waves. Block covers 32 rows; wave w -> row tile
// (w>>2), col tile (w&3); each wave computes one 16x16 tile with 16
// V_WMMA_F32_16X16X4_F32 steps over K=64. Operands staged in LDS via
// GLOBAL_LOAD_ASYNC_TO_LDS_B128 when available.
// ---------------------------------------------------------------------------
__global__ __launch_bounds__(256)
void gemm64_wmma(const float* __restrict__ A, const float* __restrict__ B,
                 const float* __restrict__ C0, float* __restrict__ D,
                 int M, int act_elu)
{
  __shared__ float sA[32 * 64];
  __shared__ float sB[64 * 64];
  const int t = threadIdx.x;
  const int row0 = blockIdx.x * 32;

#if HAVE_ASYNC_LDS
  // B: 4096 floats = 1024 x 16B chunks; A tile: 2048 floats = 512 chunks.
  for (int i = t; i < (64 * 64) / 4; i += 256)
    async_cp16((const float4*)B + i, (float4*)sB + i);
  for (int i = t; i < (32 * 64) / 4; i += 256) {
    int r = i >> 4;           // 16 float4 per row
    int c4 = i & 15;
    int gr = row0 + r; if (gr >= M) gr = M - 1;   // clamp (stores are guarded)
    async_cp16((const float4*)(A + (size_t)gr * 64) + c4, (float4*)sA + i);
  }
  wait_async0();
#else
  for (int i = t; i < (64 * 64) / 4; i += 256)
    ((float4*)sB)[i] = ((const float4*)B)[i];
  for (int i = t; i < (32 * 64) / 4; i += 256) {
    int r = i >> 4;
    int c4 = i & 15;
    int gr = row0 + r; if (gr >= M) gr = M - 1;
    ((float4*)sA)[i] = ((const float4*)(A + (size_t)gr * 64))[c4];
  }
#endif
  __syncthreads();

  const int wave = t >> 5, lane = t & 31;
  const int rt = wave >> 2, ct = wave & 3;
  const int lrow = lane & 15, hi = lane >> 4;
  const int gcol = ct * 16 + lrow;

  v8f c = {};
  if (C0 != nullptr) {                       // uniform branch
#pragma unroll
    for (int v = 0; v < 8; ++v) {
      int gr = row0 + rt * 16 + v + 8 * hi;
      if (gr >= M) gr = M - 1;
      c[v] = C0[(size_t)gr * 64 + gcol];
    }
  }

  const float* Ar = sA + (rt * 16 + lrow) * 64;
#if __has_builtin(__builtin_amdgcn_wmma_f32_16x16x4_f32)
#pragma unroll
  for (int k0 = 0; k0 < 64; k0 += 4) {
    v2f a, b;
    a.x = Ar[k0 + 2 * hi];                     // A frag: lane<16 K={0,1}, else K={2,3}
    a.y = Ar[k0 + 2 * hi + 1];
    b.x = sB[(k0 + 2 * hi) * 64 + gcol];       // B frag: K = v + 2*hi, N = lane&15
    b.y = sB[(k0 + 2 * hi + 1) * 64 + gcol];
    c = __builtin_amdgcn_wmma_f32_16x16x4_f32(false, a, false, b,
                                              (short)0, c, false, false);
  }
#else
  // scalar fallback producing identical lane layout (should not be needed)
  for (int v = 0; v < 8; ++v) {
    const float* Arow = sA + (rt * 16 + v + 8 * hi) * 64;
    float acc = c[v];
    for (int k = 0; k < 64; ++k) acc += Arow[k] * sB[k * 64 + gcol];
    c[v] = acc;
  }
#endif

#pragma unroll
  for (int v = 0; v < 8; ++v) {
    int gr = row0 + rt * 16 + v + 8 * hi;
    if (gr < M) {
      float x = c[v];
      if (act_elu) x = (x > 0.0f) ? x : (__expf(x) - 1.0f);
      D[(size_t)gr * 64 + gcol] = x;
    }
  }
}

// ---------------------------------------------------------------------------
// Fused per-node attention logits for src/dst roles:
//   s0[n*4+h] = dot(feat[n,h,:], a0[h,:]),  s1[n*4+h] = dot(feat[n,h,:], a1[h,:])
// ---------------------------------------------------------------------------
__global__ void node_scores2(const float* __restrict__ feat,
                             const float* __restrict__ a0,
                             const float* __restrict__ a1,
                             float* __restrict__ s0, float* __restrict__ s1,
                             int n_items)
{
  int idx = blockIdx.x * blockDim.x + threadIdx.x;
  if (idx >= n_items * HEADS) return;
  int n = idx >> 2, h = idx & 3;
  const float* f = feat + (size_t)n * HID + h * DH;
  const float* av0 = a0 + h * DH;
  const float* av1 = a1 + h * DH;
  float acc0 = 0.f, acc1 = 0.f;
#pragma unroll
  for (int d = 0; d < DH; ++d) {
    float x = f[d];
    acc0 += x * av0[d];
    acc1 += x * av1[d];
  }
  s0[idx] = acc0;
  s1[idx] = acc1;
}

__global__ void node_scores(const float* __restrict__ feat,
                            const float* __restrict__ a,
                            float* __restrict__ s, int n_items)
{
  int idx = blockIdx.x * blockDim.x + threadIdx.x;
  if (idx >= n_items * HEADS) return;
  int n = idx >> 2, h = idx & 3;
  const float* f = feat + (size_t)n * HID + h * DH;
  const float* av = a + h * DH;
  float acc = 0.f;
#pragma unroll
  for (int d = 0; d < DH; ++d) acc += f[d] * av[d];
  s[idx] = acc;
}

__device__ __forceinline__ float lrelu(float x) {
  return x >= 0.f ? x : SLOPE * x;
}

__global__ void edge_scores(const int* __restrict__ src, const int* __restrict__ dst,
                            const int* __restrict__ rel,
                            const float* __restrict__ s_src,
                            const float* __restrict__ s_dst,
                            const float* __restrict__ s_rel,
                            float* __restrict__ sc, int nE)
{
  int e = blockIdx.x * blockDim.x + threadIdx.x;
  if (e >= nE) return;
  int s = src[e], d = dst[e], r = rel[e];
  float4 a = ((const float4*)s_src)[s];
  float4 b = ((const float4*)s_dst)[d];
  float4 c = ((const float4*)s_rel)[r];
  float4 o;
  o.x = lrelu(a.x + b.x + c.x);
  o.y = lrelu(a.y + b.y + c.y);
  o.z = lrelu(a.z + b.z + c.z);
  o.w = lrelu(a.w + b.w + c.w);
  ((float4*)sc)[e] = o;
}

// ------------------------------- CSR by dst --------------------------------
__global__ void fill_u32(unsigned* __restrict__ p, unsigned v, int n)
{
  int i = blockIdx.x * blockDim.x + threadIdx.x;
  if (i < n) p[i] = v;
}

__global__ void count_dst(const int* __restrict__ dst, unsigned* __restrict__ cnt, int nE)
{
  int e = blockIdx.x * blockDim.x + threadIdx.x;
  if (e < nE) atomicAdd(&cnt[dst[e]], 1u);
}

__global__ void scan_offsets(const unsigned* __restrict__ cnt,
                             unsigned* __restrict__ off, int n)
{
  __shared__ unsigned sh[256];
  const int t = threadIdx.x;
  unsigned carry = 0;
  for (int base = 0; base < n; base += 256) {
    unsigned v = (base + t < n) ? cnt[base + t] : 0u;
    sh[t] = v;
    __syncthreads();
    for (int o = 1; o < 256; o <<= 1) {
      unsigned x = (t >= o) ? sh[t - o] : 0u;
      __syncthreads();
      sh[t] += x;
      __syncthreads();
    }
    if (base + t < n) off[base + t] = carry + sh[t] - v;   // exclusive
    unsigned total = sh[255];
    __syncthreads();
    carry += total;
  }
  if (t == 0) off[n] = carry;
}

__global__ void scatter_edges(const int* __restrict__ dst,
                              const unsigned* __restrict__ off,
                              unsigned* __restrict__ cur,
                              int* __restrict__ eidx, int nE)
{
  int e = blockIdx.x * blockDim.x + threadIdx.x;
  if (e >= nE) return;
  int d = dst[e];
  unsigned p = off[d] + atomicAdd(&cur[d], 1u);
  eidx[p] = e;
}

// ---------- per (node,head): top-10 threshold, max, softmax denom ----------
__global__ void seg_stats(const unsigned* __restrict__ off,
                          const int* __restrict__ eidx,
                          const float* __restrict__ sc,
                          float* __restrict__ thr, float* __restrict__ mx,
                          float* __restrict__ den, int nN)
{
  int idx = blockIdx.x * blockDim.x + threadIdx.x;
  if (idx >= nN * HEADS) return;
  int n = idx >> 2, h = idx & 3;
  unsigned b = off[n], e = off[n + 1];
  float top[TOPK];
#pragma unroll
  for (int j = 0; j < TOPK; ++j) top[j] = -3.402823466e38f;
  for (unsigned p = b; p < e; ++p) {
    float s = sc[(size_t)eidx[p] * HEADS + h];
#pragma unroll
    for (int j = 0; j < TOPK; ++j) {          // branch-free sorted insert
      float hi2 = fmaxf(top[j], s);
      s = fminf(top[j], s);
      top[j] = hi2;
    }
  }
  float th = top[TOPK - 1];                   // -inf if degree < TOPK (keep all)
  float m = (e > b) ? top[0] : 0.0f;          // nodes without edges -> m = 0
  float dsum = 0.f;
  for (unsigned p = b; p < e; ++p) {
    float s = sc[(size_t)eidx[p] * HEADS + h];
    if (s >= th) dsum += __expf(s - m);
  }
  thr[idx] = th; mx[idx] = m; den[idx] = dsum;
}

__global__ void edge_attn(const int* __restrict__ dst, const float* __restrict__ sc,
                          const float* __restrict__ thr, const float* __restrict__ mx,
                          const float* __restrict__ den, float* __restrict__ attn, int nE)
{
  int e = blockIdx.x * blockDim.x + threadIdx.x;
  if (e >= nE) return;
  int d = dst[e];
  float4 s = ((const float4*)sc)[e];
  float4 t = ((const float4*)thr)[d];
  float4 m = ((const float4*)mx)[d];
  float4 dn = ((const float4*)den)[d];
  float4 o;
  o.x = (s.x >= t.x) ? __expf(s.x - m.x) / (dn.x + 1e-16f) : 0.f;
  o.y = (s.y >= t.y) ? __expf(s.y - m.y) / (dn.y + 1e-16f) : 0.f;
  o.z = (s.z >= t.z) ? __expf(s.z - m.z) / (dn.z + 1e-16f) : 0.f;
  o.w = (s.w >= t.w) ? __expf(s.w - m.w) / (dn.w + 1e-16f) : 0.f;
  ((float4*)attn)[e] = o;
}

// ------------- one diffusion hop: hout = 0.85*SpMM + 0.15*feat0 ------------
// 4 nodes per 256-thread block; one-edge-ahead prefetch of the gathered
// source row (lowers to global_prefetch_b8 on gfx1250).
__global__ __launch_bounds__(256)
void diffuse(const unsigned* __restrict__ off, const int* __restrict__ eidx,
             const int* __restrict__ src, const float* __restrict__ attn,
             const float* __restrict__ hin, const float* __restrict__ feat0,
             float* __restrict__ hout, int nN)
{
  int n = blockIdx.x * 4 + (threadIdx.x >> 6);
  if (n >= nN) return;
  int f = threadIdx.x & 63;
  int h = (f >> 4) & 3;
  unsigned b = off[n], e = off[n + 1];
  float acc = 0.f;
  for (unsigned p = b; p < e; ++p) {
    int id = eidx[p];
    if (p + 1 < e) {
      int id2 = eidx[p + 1];
      __builtin_prefetch(hin + (size_t)src[id2] * HID + f, 0, 3);
    }
    float a = attn[(size_t)id * HEADS + h];
    acc += a * hin[(size_t)src[id] * HID + f];
  }
  hout[(size_t)n * HID + f] = (1.f - ALPHA) * acc + ALPHA * feat0[(size_t)n * HID + f];
}

// ---------------------------------------------------------------------------
extern "C" void kernel_launch(void* const* d_in, const int* in_sizes, int n_in,
                              void* d_out, int out_size, void* d_ws, size_t ws_size,
                              hipStream_t stream)
{
  (void)n_in; (void)out_size; (void)ws_size;
  const float* ent    = (const float*)d_in[0];
  const float* rel    = (const float*)d_in[1];
  const int*   esrc   = (const int*)d_in[2];
  const int*   edst   = (const int*)d_in[3];
  const int*   erel   = (const int*)d_in[4];
  const float* W_ent  = (const float*)d_in[5];
  const float* W_rel  = (const float*)d_in[6];
  const float* attn_a = (const float*)d_in[7];
  const float* res_W  = (const float*)d_in[8];
  const float* fc_rel = (const float*)d_in[9];

  const int N = in_sizes[0] / HID;
  const int R = in_sizes[1] / HID;
  const int E = in_sizes[2];

  // workspace carve-out (~100 MB), everything L2-resident on MI455X (192 MB)
  char* p = (char*)d_ws;
  auto take = [&](size_t bytes) -> void* {
    void* q = (void*)p;
    p += (bytes + 255) & ~(size_t)255;
    return q;
  };
  float*    feat0 = (float*)take((size_t)N * HID * 4);
  float*    relp  = (float*)take((size_t)R * HID * 4);
  float*    s_src = (float*)take((size_t)N * HEADS * 4);
  float*    s_dst = (float*)take((size_t)N * HEADS * 4);
  float*    s_rel = (float*)take((size_t)R * HEADS * 4);
  float*    sc    = (float*)take((size_t)E * HEADS * 4);
  float*    attn  = (float*)take((size_t)E * HEADS * 4);
  float*    thr   = (float*)take((size_t)N * HEADS * 4);
  float*    mx    = (float*)take((size_t)N * HEADS * 4);
  float*    den   = (float*)take((size_t)N * HEADS * 4);
  unsigned* cnt   = (unsigned*)take((size_t)N * 4);
  unsigned* cur   = (unsigned*)take((size_t)N * 4);
  unsigned* off   = (unsigned*)take((size_t)(N + 1) * 4);
  int*      eidx  = (int*)take((size_t)E * 4);
  float*    hcA   = (float*)take((size_t)N * HID * 4);
  float*    hcB   = (float*)take((size_t)N * HID * 4);
  float*    h1    = (float*)take((size_t)N * HID * 4);

  dim3 b256(256);
  auto g = [](int n) { return dim3((unsigned)((n + 255) / 256)); };

  // CSR by destination -- topology is layer-invariant, build once
  fill_u32<<<g(N), b256, 0, stream>>>(cnt, 0u, N);
  fill_u32<<<g(N), b256, 0, stream>>>(cur, 0u, N);
  count_dst<<<g(E), b256, 0, stream>>>(edst, cnt, E);
  scan_offsets<<<1, 256, 0, stream>>>(cnt, off, N);
  scatter_edges<<<g(E), b256, 0, stream>>>(edst, off, cur, eidx, E);

  const float* h = ent;
  float* out = (float*)d_out;
  for (int l = 0; l < 2; ++l) {
    const float* We = W_ent + (size_t)l * HID * HID;
    const float* Wr = W_rel + (size_t)l * HID * HID;
    const float* Rw = res_W + (size_t)l * HID * HID;
    const float* a0 = attn_a + (size_t)(l * 3 + 0) * HID;
    const float* a1 = attn_a + (size_t)(l * 3 + 1) * HID;
    const float* a2 = attn_a + (size_t)(l * 3 + 2) * HID;

    gemm64_wmma<<<dim3((N + 31) / 32), b256, 0, stream>>>(h, We, nullptr, feat0, N, 0);
    gemm64_wmma<<<dim3((R + 31) / 32), b256, 0, stream>>>(rel, Wr, nullptr, relp, R, 0);

    node_scores2<<<g(N * HEADS), b256, 0, stream>>>(feat0, a0, a1, s_src, s_dst, N);
    node_scores<<<g(R * HEADS), b256, 0, stream>>>(relp, a2, s_rel, R);

    edge_scores<<<g(E), b256, 0, stream>>>(esrc, edst, erel, s_src, s_dst, s_rel, sc, E);
    seg_stats<<<g(N * HEADS), b256, 0, stream>>>(off, eidx, sc, thr, mx, den, N);
    edge_attn<<<g(E), b256, 0, stream>>>(edst, sc, thr, mx, den, attn, E);

    const float* hin = feat0;
    float* bufs[2] = {hcA, hcB};
    for (int hop = 0; hop < 3; ++hop) {
      float* ho = bufs[hop & 1];
      diffuse<<<dim3((unsigned)((N + 3) / 4)), b256, 0, stream>>>(off, eidx, esrc, attn,
                                                                  hin, feat0, ho, N);
      hin = ho;
    }

    float* dstbuf = (l == 0) ? h1 : out;
    gemm64_wmma<<<dim3((N + 31) / 32), b256, 0, stream>>>(h, Rw, hin, dstbuf, N, 1);
    h = dstbuf;
  }

  // h_r = rel_emb @ fc_rel_w -> rows [N, N+R) of the output
  gemm64_wmma<<<dim3((R + 31) / 32), b256, 0, stream>>>(rel, fc_rel, nullptr,
                                                        out + (size_t)N * HID, R, 0);
}